// GCNScatterGather_4629974745747
// MI455X (gfx1250) — compile-verified
//
#include <hip/hip_runtime.h>

// ---------------------------------------------------------------------------
// GCN 2-layer for MI455X (gfx1250).  f32 WMMA (V_WMMA_F32_16X16X4_F32) for the
// dense GEMMs; float4-vectorized gather + L2 atomic scatter for aggregation.
// ---------------------------------------------------------------------------

typedef __attribute__((ext_vector_type(2))) float v2f;
typedef __attribute__((ext_vector_type(8))) float v8f;

#define NNODES 50000
#define INC    512
#define HIDC   128
#define OUTC   256
#define NEDGES 800000

// ---------------------------------------------------------------------------
// H[n][co] = sum_k X[n][k] * W[co][k]        (W is torch-layout [COUT, K])
// One wave computes one 16x16 output tile; K stepped by 4 via f32 WMMA.
//
// A-fragment (16x4 f32, ISA 7.12.2): lane L holds row M=L&15, K = 2*(L>>4)+{0,1}
//   in VGPR0/VGPR1  -> contiguous 8B load from the X row.
// B-fragment (4x16): lanes 0-15 hold K rows {0,1}, lanes 16-31 hold {2,3},
//   column N = L&15 -> contiguous 8B load from W row n (since (W^T)[k][n]=W[n][k]).
// C/D: lane column = L&15, VGPR r = row r + 8*(L>>4).
// ---------------------------------------------------------------------------
template <int K, int COUT>
__global__ __launch_bounds__(128)
void gcn_gemm_wmma(const float* __restrict__ X,
                   const float* __restrict__ W,
                   float* __restrict__ H)
{
    constexpr int tilesN     = COUT / 16;
    constexpr int tilesTotal = (NNODES / 16) * tilesN;

    const int wave = threadIdx.x >> 5;
    const int lane = threadIdx.x & 31;
    const int tile = blockIdx.x * 4 + wave;
    if (tile >= tilesTotal) return;              // wave-uniform: EXEC stays all-1s

    const int tm   = tile / tilesN;
    const int tn   = tile % tilesN;
    const int half = lane >> 4;                  // 0: K pair {0,1}, 1: K pair {2,3}
    const int l15  = lane & 15;

    const float* __restrict__ xrow = X + (size_t)(tm * 16 + l15) * K + 2 * half;
    const float* __restrict__ wrow = W + (size_t)(tn * 16 + l15) * K + 2 * half;

    v8f acc = {};
#pragma unroll 8
    for (int k = 0; k < K; k += 4) {
        v2f a = *(const v2f*)(xrow + k);
        v2f b = *(const v2f*)(wrow + k);
        // 8 args: (neg_a, A, neg_b, B, c_mod, C, reuse_a, reuse_b)
        acc = __builtin_amdgcn_wmma_f32_16x16x4_f32(
            false, a, false, b, (short)0, acc, false, false);
    }

    float* __restrict__ out = H + (size_t)(tm * 16) * COUT + tn * 16 + l15;
#pragma unroll
    for (int r = 0; r < 8; ++r)
        out[(size_t)(r + 8 * half) * COUT] = acc[r];
}

// ---------------------------------------------------------------------------
// Elementwise helpers
// ---------------------------------------------------------------------------
__global__ void zero_init_kernel(float* __restrict__ p, size_t total)
{
    size_t t = (size_t)blockIdx.x * blockDim.x + threadIdx.x;
    if (t < total) p[t] = 0.0f;
}

__global__ void set_bias_kernel(float* __restrict__ p, const float* __restrict__ b,
                                int cmask, size_t total)
{
    size_t t = (size_t)blockIdx.x * blockDim.x + threadIdx.x;
    if (t < total) p[t] = b[t & cmask];
}

__global__ void bias_relu_kernel(float* __restrict__ p, const float* __restrict__ b,
                                 int cmask, size_t total)
{
    size_t t = (size_t)blockIdx.x * blockDim.x + threadIdx.x;
    if (t < total) {
        float v = p[t] + b[t & cmask];
        p[t] = v > 0.0f ? v : 0.0f;
    }
}

// ---------------------------------------------------------------------------
// Scatter-add: AGG[dst[e]][:] += H[src[e]][:]
// Consecutive lanes handle consecutive float4 channel groups of one edge
// (C=128 -> 1 wave/edge, C=256 -> 2 waves/edge): coalesced 16B gathers,
// 4 global f32 atomics per lane into L2.
// ---------------------------------------------------------------------------
template <int C>
__global__ __launch_bounds__(256)
void scatter_add_kernel(const float* __restrict__ H,
                        const int* __restrict__ eidx,   // [2, E] flat: src then dst
                        float* __restrict__ AGG)
{
    constexpr int G = C / 4;                     // float4 groups per row (pow2)
    long long t = (long long)blockIdx.x * blockDim.x + threadIdx.x;
    long long e = t >> (__builtin_ctz(G));       // t / G
    if (e >= NEDGES) return;
    int g = (int)(t & (G - 1));                  // t % G

    int s = eidx[e];
    int d = eidx[NEDGES + e];

    const float4 v = *(const float4*)(H + (size_t)s * C + g * 4);
    float* p = AGG + (size_t)d * C + g * 4;
    atomicAdd(p + 0, v.x);
    atomicAdd(p + 1, v.y);
    atomicAdd(p + 2, v.z);
    atomicAdd(p + 3, v.w);
}

// ---------------------------------------------------------------------------
extern "C" void kernel_launch(void* const* d_in, const int* in_sizes, int n_in,
                              void* d_out, int out_size, void* d_ws, size_t ws_size,
                              hipStream_t stream)
{
    const float* x  = (const float*)d_in[0];
    const int*   ei = (const int*)  d_in[1];     // [2, E] flat (src row, dst row)
    const float* w1 = (const float*)d_in[2];
    const float* b1 = (const float*)d_in[3];
    const float* w2 = (const float*)d_in[4];
    const float* b2 = (const float*)d_in[5];
    float* out = (float*)d_out;

    // Workspace layout (floats): h1 [N*128] | agg1/h2 [N*128] | g2 [N*256]
    float* h1   = (float*)d_ws;
    float* agg1 = h1   + (size_t)NNODES * HIDC;
    float* g2   = agg1 + (size_t)NNODES * HIDC;

    const size_t n_hid = (size_t)NNODES * HIDC;   // 6.4M
    const size_t n_out = (size_t)NNODES * OUTC;   // 12.8M

    // ---- Layer 1 GEMM: h1 = x @ w1^T  (f32 WMMA) ----
    {
        constexpr int tiles = (NNODES / 16) * (HIDC / 16);   // 25000 waves
        gcn_gemm_wmma<INC, HIDC><<<tiles / 4, 128, 0, stream>>>(x, w1, h1);
    }

    // ---- agg1 = 0; agg1[dst] += h1[src] ----
    zero_init_kernel<<<(int)((n_hid + 255) / 256), 256, 0, stream>>>(agg1, n_hid);
    {
        long long threads = (long long)NEDGES * (HIDC / 4);  // 25.6M
        scatter_add_kernel<HIDC><<<(int)((threads + 255) / 256), 256, 0, stream>>>(
            h1, ei, agg1);
    }

    // ---- h2 = relu(agg1 + b1)  (in place) ----
    bias_relu_kernel<<<(int)((n_hid + 255) / 256), 256, 0, stream>>>(
        agg1, b1, HIDC - 1, n_hid);

    // ---- Layer 2 GEMM: g2 = h2 @ w2^T  (f32 WMMA) ----
    {
        constexpr int tiles = (NNODES / 16) * (OUTC / 16);   // 50000 waves
        gcn_gemm_wmma<HIDC, OUTC><<<tiles / 4, 128, 0, stream>>>(agg1, w2, g2);
    }

    // ---- out = b2 (broadcast); out[dst] += g2[src] ----
    set_bias_kernel<<<(int)((n_out + 255) / 256), 256, 0, stream>>>(
        out, b2, OUTC - 1, n_out);
    {
        long long threads = (long long)NEDGES * (OUTC / 4);  // 51.2M
        scatter_add_kernel<OUTC><<<(int)((threads + 255) / 256), 256, 0, stream>>>(
            g2, ei, out);
    }
}